// MyRegLoss_23759759082228
// MI455X (gfx1250) — compile-verified
//
#include <hip/hip_runtime.h>

// ---------------------------------------------------------------------------
// Masked smooth-L1 reduction, MI455X (gfx1250, wave32).
//
// Bandwidth-bound: 268 MB of f32 reads -> ~11.5 us floor @ 23.3 TB/s.
// Strategy: float4 (b128) coalesced grid-stride streaming, two f32
// accumulators per lane, then a wave32 reduction done with
// V_WMMA_F32_16X16X4_F32 (A = 16x4 matrix of the wave's 64 partial sums,
// B = ones => D rows are exact f32 row-sums). Two-pass deterministic
// block reduction (no float atomics) so graph replays are bit-identical.
// ---------------------------------------------------------------------------

#define NBLOCKS  2048
#define NTHREADS 256
#define NWAVES   (NTHREADS / 32)

typedef __attribute__((ext_vector_type(2))) float v2f;
typedef __attribute__((ext_vector_type(8))) float v8f;

__device__ __forceinline__ float sl1_masked(float o, float t) {
    float d   = fabsf(o - t);
    float sl1 = (d < 1.0f) ? (0.5f * d * d) : (d - 0.5f);
    return (t != 0.0f) ? sl1 : 0.0f;
}

// Wave32 reduction of (acc0, acc1) per lane -> wave total in every lane.
// A-matrix (16x4 f32, 2 VGPRs): lanes 0-15 hold (K=0,K=1) for M=lane,
// lanes 16-31 hold (K=2,K=3) for M=lane-16 -- i.e. exactly {acc0, acc1}.
// With B = ones, D[m][n] = sum_k A[m][k]. D VGPR r: lanes 0-15 = row r,
// lanes 16-31 = row r+8, so sum of a lane's 8 D regs = half-total; one
// shfl_xor(16) folds the halves. All f32, order fixed by HW => deterministic.
__device__ __forceinline__ float wave_reduce(float acc0, float acc1) {
#if __has_builtin(__builtin_amdgcn_wmma_f32_16x16x4_f32)
    v2f a; a[0] = acc0; a[1] = acc1;
    v2f b; b[0] = 1.0f; b[1] = 1.0f;
    v8f c = {};
    c = __builtin_amdgcn_wmma_f32_16x16x4_f32(
            /*neg_a=*/false, a, /*neg_b=*/false, b,
            /*c_mod=*/(short)0, c, /*reuse_a=*/false, /*reuse_b=*/false);
    float s = ((c[0] + c[1]) + (c[2] + c[3])) + ((c[4] + c[5]) + (c[6] + c[7]));
    s += __shfl_xor(s, 16, 32);
    return s;
#else
    float s = acc0 + acc1;
    #pragma unroll
    for (int off = 16; off > 0; off >>= 1) s += __shfl_xor(s, off, 32);
    return s;
#endif
}

__global__ __launch_bounds__(NTHREADS)
void sl1_partial_kernel(const float* __restrict__ o,
                        const float* __restrict__ t,
                        float* __restrict__ partial,
                        int n) {
    const int n4 = n >> 2;
    const float4* __restrict__ o4 = reinterpret_cast<const float4*>(o);
    const float4* __restrict__ t4 = reinterpret_cast<const float4*>(t);

    const int tid    = (int)(blockIdx.x * NTHREADS + threadIdx.x);
    const int stride = (int)(gridDim.x * NTHREADS);

    float acc0 = 0.0f, acc1 = 0.0f;
    int i = tid;
    // two streams per trip -> two independent accumulators (ILP), b128 loads
    for (; i + stride < n4; i += 2 * stride) {
        float4 a = o4[i];
        float4 b = t4[i];
        acc0 += (sl1_masked(a.x, b.x) + sl1_masked(a.y, b.y)) +
                (sl1_masked(a.z, b.z) + sl1_masked(a.w, b.w));
        float4 a2 = o4[i + stride];
        float4 b2 = t4[i + stride];
        acc1 += (sl1_masked(a2.x, b2.x) + sl1_masked(a2.y, b2.y)) +
                (sl1_masked(a2.z, b2.z) + sl1_masked(a2.w, b2.w));
    }
    if (i < n4) {
        float4 a = o4[i];
        float4 b = t4[i];
        acc0 += (sl1_masked(a.x, b.x) + sl1_masked(a.y, b.y)) +
                (sl1_masked(a.z, b.z) + sl1_masked(a.w, b.w));
    }
    // scalar tail (n not multiple of 4) handled once, deterministically
    if (tid == 0) {
        for (int k = (n4 << 2); k < n; ++k) acc0 += sl1_masked(o[k], t[k]);
    }

    // control flow reconverged: EXEC is all-1s here (WMMA requirement)
    const float wsum = wave_reduce(acc0, acc1);

    __shared__ float ws[NWAVES];
    if ((threadIdx.x & 31) == 0) ws[threadIdx.x >> 5] = wsum;
    __syncthreads();
    if (threadIdx.x == 0) {
        float s = 0.0f;
        #pragma unroll
        for (int k = 0; k < NWAVES; ++k) s += ws[k];   // fixed order
        partial[blockIdx.x] = s;
    }
}

__global__ __launch_bounds__(NTHREADS)
void sl1_final_kernel(const float* __restrict__ partial,
                      float* __restrict__ out) {
    float acc0 = 0.0f, acc1 = 0.0f;
    #pragma unroll
    for (int i = (int)threadIdx.x; i < NBLOCKS; i += 2 * NTHREADS) {
        acc0 += partial[i];
        if (i + NTHREADS < NBLOCKS) acc1 += partial[i + NTHREADS];
    }

    const float wsum = wave_reduce(acc0, acc1);

    __shared__ float ws[NWAVES];
    if ((threadIdx.x & 31) == 0) ws[threadIdx.x >> 5] = wsum;
    __syncthreads();
    if (threadIdx.x == 0) {
        float s = 0.0f;
        #pragma unroll
        for (int k = 0; k < NWAVES; ++k) s += ws[k];   // fixed order
        out[0] = s;
    }
}

extern "C" void kernel_launch(void* const* d_in, const int* in_sizes, int n_in,
                              void* d_out, int out_size, void* d_ws, size_t ws_size,
                              hipStream_t stream) {
    (void)n_in; (void)out_size; (void)ws_size;
    const float* o = (const float*)d_in[0];   // "out"    (f32)
    const float* t = (const float*)d_in[1];   // "target" (f32)
    float* result  = (float*)d_out;           // scalar f32
    float* partial = (float*)d_ws;            // NBLOCKS f32 partials (8 KB)
    const int n = in_sizes[0];

    sl1_partial_kernel<<<NBLOCKS, NTHREADS, 0, stream>>>(o, t, partial, n);
    sl1_final_kernel<<<1, NTHREADS, 0, stream>>>(partial, result);
}